// TripletLoss_13030930776533
// MI455X (gfx1250) — compile-verified
//
#include <hip/hip_runtime.h>
#include <hip/hip_bf16.h>

typedef __attribute__((ext_vector_type(2))) float v2f;
typedef __attribute__((ext_vector_type(8))) float v8f;

#define N 384
#define D 128
#define MARGIN_F 1.0f
#define EPS_F 1e-16f

// ---------------------------------------------------------------------------
// Kernel 1: per-row squared norms  sq[t] = sum_k x[t][k]^2
// ---------------------------------------------------------------------------
__global__ __launch_bounds__(128) void norms_kernel(const float* __restrict__ x,
                                                    float* __restrict__ sq) {
    int t = blockIdx.x * blockDim.x + threadIdx.x;
    if (t < N) {
        const float* row = x + t * D;
        float s = 0.0f;
#pragma unroll 8
        for (int k = 0; k < D; ++k) s += row[k] * row[k];
        sq[t] = s;
    }
}

// ---------------------------------------------------------------------------
// Kernel 2: dist[i][j] via WMMA f32 Gram matrix + fused distance epilogue.
// One wave (32 lanes) per 16x16 output tile; K=128 in steps of 4 with
// V_WMMA_F32_16X16X4_F32.
//
// A-matrix 16x4 f32 layout (ISA 7.12.2): lanes 0-15 M=0..15 holding K={0,1},
// lanes 16-31 same M holding K={2,3}  -> each lane loads a contiguous float2.
// B-matrix 4x16 mirrored with N = lane%16 (B = X^T tile, rows contiguous in X).
// C/D: VGPR r, lane L -> M = r + 8*(L/16), N = L%16.
// ---------------------------------------------------------------------------
__global__ __launch_bounds__(32) void dist_wmma_kernel(const float* __restrict__ x,
                                                       const float* __restrict__ sq,
                                                       float* __restrict__ dist) {
    const int bi = blockIdx.x;
    const int bj = blockIdx.y;
    const int lane = threadIdx.x & 31;
    const int half = lane >> 4;   // 0: lanes 0-15, 1: lanes 16-31
    const int l16  = lane & 15;

    const int arow = bi * 16 + l16;   // A-matrix row (M) for this lane
    const int bcol = bj * 16 + l16;   // B-matrix column (N) for this lane
    const int koff = half * 2;        // which K-pair this lane half supplies

    v8f c = {};
#pragma unroll 4
    for (int k = 0; k < D; k += 4) {
        v2f a = *(const v2f*)(x + arow * D + k + koff);
        v2f b = *(const v2f*)(x + bcol * D + k + koff);
        // D = A x B + C  (16x16 f32 accumulator in 8 VGPRs)
        c = __builtin_amdgcn_wmma_f32_16x16x4_f32(
                /*neg_a=*/false, a, /*neg_b=*/false, b,
                /*c_mod=*/(short)0, c, /*reuse_a=*/false, /*reuse_b=*/false);
    }

    // Epilogue on accumulator registers: d2 = sq_i + sq_j - 2*gram; safe sqrt.
    const int j = bj * 16 + l16;
    const float sqj = sq[j];
#pragma unroll
    for (int r = 0; r < 8; ++r) {
        int i = bi * 16 + r + half * 8;
        float d2 = sq[i] + sqj - 2.0f * c[r];
        d2 = fmaxf(d2, 0.0f);
        dist[i * N + j] = (d2 > 0.0f) ? __builtin_sqrtf(d2) : 0.0f;
    }
}

// ---------------------------------------------------------------------------
// Kernel 3: masked triplet scan, positives-compacted.
// One block per anchor i. mask(i,j,k) = (lab_j==lab_i && j!=i && lab_k!=lab_i)
// (other distinctness terms are implied). Instead of sweeping the full
// 384x384 (j,k) plane (~97% masked off), deterministically compact the
// positive-j list (~11 entries) and sweep only npos*384 pairs.
// num_valid = npos * nneg analytically. Per-anchor partials written out.
// ---------------------------------------------------------------------------
__global__ __launch_bounds__(256) void triplet_kernel(const float* __restrict__ dist,
                                                      const int* __restrict__ labels,
                                                      float* __restrict__ partials) {
    __shared__ float sd[N];
    __shared__ int   sl[N];
    __shared__ int   plist[N];
    __shared__ int   s_np;
    __shared__ int   s_nn;
    __shared__ float r0[256];
    __shared__ float r1[256];

    const int i   = blockIdx.x;
    const int tid = threadIdx.x;

    for (int t = tid; t < N; t += 256) {
        sd[t] = dist[i * N + t];
        sl[t] = labels[t];
    }
    __syncthreads();

    const int li = sl[i];

    // Deterministic compaction of positives (fixed order => bit-stable sums).
    if (tid == 0) {
        int np = 0, nn = 0;
        for (int t = 0; t < N; ++t) {
            if (sl[t] == li) {
                if (t != i) plist[np++] = t;
            } else {
                ++nn;
            }
        }
        s_np = np;
        s_nn = nn;
    }
    __syncthreads();

    const int npos  = s_np;
    const int nneg  = s_nn;
    const int total = npos * N;

    float    sum = 0.0f;
    unsigned pos = 0u;

    for (int idx = tid; idx < total; idx += 256) {
        int p = idx / N;
        int k = idx - p * N;
        if (sl[k] != li) {
            float t = sd[plist[p]] - sd[k] + MARGIN_F;
            if (t > 0.0f) {
                sum += t;
                if (t > EPS_F) ++pos;
            }
        }
    }

    r0[tid] = sum;
    r1[tid] = (float)pos;
    __syncthreads();
#pragma unroll
    for (int s = 128; s > 0; s >>= 1) {
        if (tid < s) {
            r0[tid] += r0[tid + s];
            r1[tid] += r1[tid + s];
        }
        __syncthreads();
    }
    if (tid == 0) {
        partials[i * 3 + 0] = r0[0];
        partials[i * 3 + 1] = r1[0];
        partials[i * 3 + 2] = (float)(npos * nneg);   // num_valid for anchor i
    }
}

// ---------------------------------------------------------------------------
// Kernel 4: deterministic final reduction over 384 partials -> (loss, frac).
// ---------------------------------------------------------------------------
__global__ void finalize_kernel(const float* __restrict__ partials,
                                float* __restrict__ out) {
    if (threadIdx.x == 0 && blockIdx.x == 0) {
        float s = 0.0f, p = 0.0f, v = 0.0f;
        for (int i = 0; i < N; ++i) {
            s += partials[3 * i + 0];
            p += partials[3 * i + 1];
            v += partials[3 * i + 2];
        }
        out[0] = s / (p + EPS_F);   // loss
        out[1] = p / (v + EPS_F);   // fraction_positive
    }
}

extern "C" void kernel_launch(void* const* d_in, const int* in_sizes, int n_in,
                              void* d_out, int out_size, void* d_ws, size_t ws_size,
                              hipStream_t stream) {
    const float* x      = (const float*)d_in[0];   // embeddings [384,128] f32
    const int*   labels = (const int*)d_in[1];     // labels [384] int

    float* ws       = (float*)d_ws;
    float* dist     = ws;                 // N*N floats
    float* sq       = ws + N * N;         // N floats
    float* partials = sq + N;             // 3*N floats

    float* out = (float*)d_out;           // 2 floats: loss, fraction_positive

    norms_kernel<<<dim3((N + 127) / 128), dim3(128), 0, stream>>>(x, sq);
    dist_wmma_kernel<<<dim3(N / 16, N / 16), dim3(32), 0, stream>>>(x, sq, dist);
    triplet_kernel<<<dim3(N), dim3(256), 0, stream>>>(dist, labels, partials);
    finalize_kernel<<<dim3(1), dim3(32), 0, stream>>>(partials, out);
}